// Net_2740189135622
// MI455X (gfx1250) — compile-verified
//
#include <hip/hip_runtime.h>
#include <math.h>

typedef __attribute__((ext_vector_type(2))) float v2f;
typedef __attribute__((ext_vector_type(8))) float v8f;

#define NEG_SLOPE 0.2f

__device__ __forceinline__ unsigned fkey(float f) {
  unsigned b = __float_as_uint(f);
  return (b & 0x80000000u) ? ~b : (b | 0x80000000u);
}
__device__ __forceinline__ float fdecode(unsigned k) {
  unsigned b = (k & 0x80000000u) ? (k ^ 0x80000000u) : ~k;
  return __uint_as_float(b);
}
__device__ __forceinline__ float lrelu(float x) { return x > 0.f ? x : NEG_SLOPE * x; }

// ---------------- utility: zero workspace ----------------
__global__ void k_zero(float* __restrict__ p, size_t n) {
  size_t i = (size_t)blockIdx.x * blockDim.x + threadIdx.x;
  size_t st = (size_t)gridDim.x * blockDim.x;
  for (; i < n; i += st) p[i] = 0.f;
}

// ---------------- layer 1 GEMM: h1[N,64] = x[N,256] @ W1[256,64] ----------------
// 8 waves/block; wave w -> rowTile = blockIdx.x*2 + w/4, colTile = w%4.
__global__ __launch_bounds__(256) void k_gemm1(const float* __restrict__ x,
                                               const float* __restrict__ W,
                                               float* __restrict__ h, int N) {
  __shared__ __align__(16) float sW[256 * 64];  // 64 KB, well under 320 KB/WGP
  {
    const float4* wv = (const float4*)W;
    float4* sv = (float4*)sW;
    for (int i = threadIdx.x; i < 256 * 64 / 4; i += 256) sv[i] = wv[i];
  }
  __syncthreads();
  int wave = threadIdx.x >> 5;
  int lane = threadIdx.x & 31;  // wave32
  int rowTile = blockIdx.x * 2 + (wave >> 2);
  int colTile = wave & 3;
  if (rowTile * 16 >= N) return;
  int lm = lane & 15, half = lane >> 4;
  int row = rowTile * 16 + lm;
  if (row >= N) row = N - 1;  // clamped rows only feed OOB output rows (not stored)
  const float* xr = x + (size_t)row * 256;
  int col = colTile * 16 + lm;
  v8f c = {};
  for (int k0 = 0; k0 < 256; k0 += 4) {
    if (k0 + 32 < 256) __builtin_prefetch(xr + k0 + 32, 0, 1);  // global_prefetch_b8
    int k = k0 + 2 * half;  // A layout: lanes 0-15 K={0,1}, lanes 16-31 K={2,3}
    v2f a, b;
    a[0] = xr[k];
    a[1] = xr[k + 1];
    b[0] = sW[k * 64 + col];
    b[1] = sW[(k + 1) * 64 + col];
    c = __builtin_amdgcn_wmma_f32_16x16x4_f32(false, a, false, b, (short)0, c, false, false);
  }
  // C/D layout: VGPR v holds row v (lanes 0-15) / row 8+v (lanes 16-31), col = lm
  for (int v = 0; v < 8; ++v) {
    int ro = rowTile * 16 + 8 * half + v;
    if (ro < N) h[(size_t)ro * 64 + col] = c[v];
  }
}

// ---------------- layer 2 GEMM: h2[N,16] = f1[N,64] @ W2[64,16] ----------------
__global__ __launch_bounds__(256) void k_gemm2(const float* __restrict__ f,
                                               const float* __restrict__ W,
                                               float* __restrict__ h, int N) {
  __shared__ __align__(16) float sW[64 * 16];  // 4 KB
  ((float4*)sW)[threadIdx.x & 255] = ((const float4*)W)[threadIdx.x & 255];
  __syncthreads();
  int wave = threadIdx.x >> 5;
  int lane = threadIdx.x & 31;
  int rowTile = blockIdx.x * 8 + wave;
  if (rowTile * 16 >= N) return;
  int lm = lane & 15, half = lane >> 4;
  int row = rowTile * 16 + lm;
  if (row >= N) row = N - 1;
  const float* fr = f + (size_t)row * 64;
  v8f c = {};
  for (int k0 = 0; k0 < 64; k0 += 4) {
    int k = k0 + 2 * half;
    v2f a, b;
    a[0] = fr[k];
    a[1] = fr[k + 1];
    b[0] = sW[k * 16 + lm];
    b[1] = sW[(k + 1) * 16 + lm];
    c = __builtin_amdgcn_wmma_f32_16x16x4_f32(false, a, false, b, (short)0, c, false, false);
  }
  for (int v = 0; v < 8; ++v) {
    int ro = rowTile * 16 + 8 * half + v;
    if (ro < N) h[(size_t)ro * 16 + lm] = c[v];
  }
}

// ---------------- per-node attention logits, layer 1 (8 heads x 8 ch) ----------------
__global__ void k_alpha1(const float* __restrict__ h, const float* __restrict__ asrc,
                         const float* __restrict__ adst, float* __restrict__ as_,
                         float* __restrict__ ad_, int N) {
  int t = blockIdx.x * blockDim.x + threadIdx.x;  // t = n*8 + head
  if (t >= N * 8) return;
  int head = t & 7;
  const float* hp = h + (size_t)(t >> 3) * 64 + head * 8;
  float s = 0.f, d = 0.f;
  for (int c = 0; c < 8; ++c) {
    float v = hp[c];
    s += v * asrc[head * 8 + c];
    d += v * adst[head * 8 + c];
  }
  as_[t] = s;
  ad_[t] = d;
}

// ---------------- layer-1 edge softmax pipeline ----------------
__global__ void k_emax1(const int* __restrict__ ei, const float* __restrict__ as_,
                        const float* __restrict__ ad_, unsigned* __restrict__ emax,
                        int E, int ET) {
  int e = blockIdx.x * blockDim.x + threadIdx.x;
  if (e >= ET) return;
  int s, d;
  if (e < E) { s = ei[e]; d = ei[E + e]; } else { s = d = e - E; }
  for (int h = 0; h < 8; ++h) {
    float v = lrelu(as_[s * 8 + h] + ad_[d * 8 + h]);
    atomicMax(&emax[d * 8 + h], fkey(v));
  }
}

__global__ void k_eexp1(const int* __restrict__ ei, const float* __restrict__ as_,
                        const float* __restrict__ ad_, const unsigned* __restrict__ emax,
                        float* __restrict__ den, float* __restrict__ eexp, int E, int ET) {
  int e = blockIdx.x * blockDim.x + threadIdx.x;
  if (e >= ET) return;
  int s, d;
  if (e < E) { s = ei[e]; d = ei[E + e]; } else { s = d = e - E; }
  for (int h = 0; h < 8; ++h) {
    float v = lrelu(as_[s * 8 + h] + ad_[d * 8 + h]);
    float m = fdecode(emax[d * 8 + h]);
    float ex = expf(v - m);
    eexp[(size_t)e * 8 + h] = ex;
    atomicAdd(&den[d * 8 + h], ex);
  }
}

__global__ void k_aggr1(const int* __restrict__ ei, const float* __restrict__ eexp,
                        const float* __restrict__ den, const float* __restrict__ h1,
                        float* __restrict__ agg, int E, int ET) {
  int t = blockIdx.x * blockDim.x + threadIdx.x;  // t = e*8 + head
  if (t >= ET * 8) return;
  int e = t >> 3, head = t & 7;
  int s, d;
  if (e < E) { s = ei[e]; d = ei[E + e]; } else { s = d = e - E; }
  float a = eexp[t] / (den[d * 8 + head] + 1e-16f);
  const float* hs = h1 + (size_t)s * 64 + head * 8;
  float* ap = agg + (size_t)d * 64 + head * 8;
  for (int c = 0; c < 8; ++c) atomicAdd(&ap[c], a * hs[c]);
}

__global__ void k_feat1(const float* __restrict__ agg, const float* __restrict__ b,
                        float* __restrict__ f, int N) {
  int t = blockIdx.x * blockDim.x + threadIdx.x;
  if (t >= N * 64) return;
  float v = agg[t] + b[t & 63];
  f[t] = v > 0.f ? v : (expf(v) - 1.f);  // ELU
}

// ---------------- layer-2 (H=1, C=16) ----------------
__global__ void k_alpha2(const float* __restrict__ h2, const float* __restrict__ asrc,
                         const float* __restrict__ adst, float* __restrict__ as_,
                         float* __restrict__ ad_, int N) {
  int n = blockIdx.x * blockDim.x + threadIdx.x;
  if (n >= N) return;
  const float* hp = h2 + (size_t)n * 16;
  float s = 0.f, d = 0.f;
  for (int c = 0; c < 16; ++c) {
    float v = hp[c];
    s += v * asrc[c];
    d += v * adst[c];
  }
  as_[n] = s;
  ad_[n] = d;
}

__global__ void k_emax2(const int* __restrict__ ei, const float* __restrict__ as_,
                        const float* __restrict__ ad_, unsigned* __restrict__ emax,
                        int E, int ET) {
  int e = blockIdx.x * blockDim.x + threadIdx.x;
  if (e >= ET) return;
  int s, d;
  if (e < E) { s = ei[e]; d = ei[E + e]; } else { s = d = e - E; }
  float v = lrelu(as_[s] + ad_[d]);
  atomicMax(&emax[d], fkey(v));
}

__global__ void k_eexp2(const int* __restrict__ ei, const float* __restrict__ as_,
                        const float* __restrict__ ad_, const unsigned* __restrict__ emax,
                        float* __restrict__ den, float* __restrict__ eexp, int E, int ET) {
  int e = blockIdx.x * blockDim.x + threadIdx.x;
  if (e >= ET) return;
  int s, d;
  if (e < E) { s = ei[e]; d = ei[E + e]; } else { s = d = e - E; }
  float v = lrelu(as_[s] + ad_[d]);
  float ex = expf(v - fdecode(emax[d]));
  eexp[e] = ex;
  atomicAdd(&den[d], ex);
}

__global__ void k_aggr2(const int* __restrict__ ei, const float* __restrict__ eexp,
                        const float* __restrict__ den, const float* __restrict__ h2,
                        float* __restrict__ agg, int E, int ET) {
  int e = blockIdx.x * blockDim.x + threadIdx.x;
  if (e >= ET) return;
  int s, d;
  if (e < E) { s = ei[e]; d = ei[E + e]; } else { s = d = e - E; }
  float a = eexp[e] / (den[d] + 1e-16f);
  const float* hs = h2 + (size_t)s * 16;
  float* ap = agg + (size_t)d * 16;
  for (int c = 0; c < 16; ++c) atomicAdd(&ap[c], a * hs[c]);
}

__global__ void k_out(const float* __restrict__ agg, const float* __restrict__ b,
                      float* __restrict__ out, int N) {
  int n = blockIdx.x * blockDim.x + threadIdx.x;
  if (n >= N) return;
  float y[16];
  float m = -3.4e38f;
  for (int c = 0; c < 16; ++c) {
    y[c] = agg[(size_t)n * 16 + c] + b[c];
    m = fmaxf(m, y[c]);
  }
  float s = 0.f;
  for (int c = 0; c < 16; ++c) s += expf(y[c] - m);
  float l = m + logf(s);
  for (int c = 0; c < 16; ++c) out[(size_t)n * 16 + c] = y[c] - l;
}

// ---------------- driver ----------------
extern "C" void kernel_launch(void* const* d_in, const int* in_sizes, int n_in,
                              void* d_out, int out_size, void* d_ws, size_t ws_size,
                              hipStream_t stream) {
  const float* x    = (const float*)d_in[0];
  const int*   ei   = (const int*)d_in[1];
  const float* W1   = (const float*)d_in[2];
  const float* a_s1 = (const float*)d_in[3];
  const float* a_d1 = (const float*)d_in[4];
  const float* b1   = (const float*)d_in[5];
  const float* W2   = (const float*)d_in[6];
  const float* a_s2 = (const float*)d_in[7];
  const float* a_d2 = (const float*)d_in[8];
  const float* b2   = (const float*)d_in[9];
  float* out = (float*)d_out;

  const int N = in_sizes[0] / 256;
  const int E = in_sizes[1] / 2;
  const int ET = E + N;  // + self loops

  float* ws = (float*)d_ws;
  size_t o = 0;
  float*    h1    = ws + o; o += (size_t)N * 64;
  float*    as1   = ws + o; o += (size_t)N * 8;
  float*    ad1   = ws + o; o += (size_t)N * 8;
  unsigned* emax1 = (unsigned*)(ws + o); o += (size_t)N * 8;
  float*    den1  = ws + o; o += (size_t)N * 8;
  float*    agg1  = ws + o; o += (size_t)N * 64;
  float*    eexp1 = ws + o; o += (size_t)ET * 8;
  float*    f1    = ws + o; o += (size_t)N * 64;
  float*    h2    = ws + o; o += (size_t)N * 16;
  float*    as2   = ws + o; o += (size_t)N;
  float*    ad2   = ws + o; o += (size_t)N;
  unsigned* emax2 = (unsigned*)(ws + o); o += (size_t)N;
  float*    den2  = ws + o; o += (size_t)N;
  float*    agg2  = ws + o; o += (size_t)N * 16;
  float*    eexp2 = ws + o; o += (size_t)ET;
  size_t total = o;

  const int B = 256;
  // 0) zero all accumulators (key=0 is a valid -inf sentinel for the ordered-uint max)
  k_zero<<<2048, B, 0, stream>>>(ws, total);

  int rowTiles = (N + 15) / 16;
  // 1) h1 = x @ W1 (WMMA)
  k_gemm1<<<(rowTiles + 1) / 2, B, 0, stream>>>(x, W1, h1, N);
  // 2) attention logits per node
  k_alpha1<<<(N * 8 + B - 1) / B, B, 0, stream>>>(h1, a_s1, a_d1, as1, ad1, N);
  // 3) segment softmax over incoming edges
  k_emax1<<<(ET + B - 1) / B, B, 0, stream>>>(ei, as1, ad1, emax1, E, ET);
  k_eexp1<<<(ET + B - 1) / B, B, 0, stream>>>(ei, as1, ad1, emax1, den1, eexp1, E, ET);
  k_aggr1<<<(ET * 8 + B - 1) / B, B, 0, stream>>>(ei, eexp1, den1, h1, agg1, E, ET);
  // 4) bias + ELU
  k_feat1<<<(N * 64 + B - 1) / B, B, 0, stream>>>(agg1, b1, f1, N);
  // 5) layer 2
  k_gemm2<<<(rowTiles + 7) / 8, B, 0, stream>>>(f1, W2, h2, N);
  k_alpha2<<<(N + B - 1) / B, B, 0, stream>>>(h2, a_s2, a_d2, as2, ad2, N);
  k_emax2<<<(ET + B - 1) / B, B, 0, stream>>>(ei, as2, ad2, emax2, E, ET);
  k_eexp2<<<(ET + B - 1) / B, B, 0, stream>>>(ei, as2, ad2, emax2, den2, eexp2, E, ET);
  k_aggr2<<<(ET + B - 1) / B, B, 0, stream>>>(ei, eexp2, den2, h2, agg2, E, ET);
  // 6) log_softmax
  k_out<<<(N + B - 1) / B, B, 0, stream>>>(agg2, b2, out, N);
}